// UtilityLoss_70531952935356
// MI455X (gfx1250) — compile-verified
//
#include <hip/hip_runtime.h>

// Clang ext-vector types: __builtin_nontemporal_load rejects HIP's struct float4.
typedef float v4f __attribute__((ext_vector_type(4)));
typedef int   v4i __attribute__((ext_vector_type(4)));

#define BLOCK 256
#define WAVES (BLOCK / 32)
#define GRID_MAX 2048

// Per-element utility value u_func (reference computes sum(-u_func)).
__device__ __forceinline__ float u_elem(float p, int flag, float s, float c) {
    // m1 = 1/6, b1 = 2 ; m2 = -1/9, b2 = 1/3 ; m3 = -2/9, b3 = -4/3 ; u_fp = -0.05
    const float m1 = 1.0f / 6.0f,  b1 = 2.0f;
    const float m2 = -1.0f / 9.0f, b2 = 1.0f / 3.0f;
    const float m3 = -2.0f / 9.0f, b3 = -4.0f / 3.0f;
    const float u_fp = -0.05f;

    float d = c - (s + 6.0f);   // d = t - (t_sepsis - dt_optimal), dt_optimal=-6
    bool in_win = d <= 3.0f;    // d <= dt_late
    bool early  = d <= -6.0f;   // d <= dt_optimal

    float uTP = early ? fmaxf(m1 * d + b1, u_fp) : (m2 * d + b2);
    float uFN = early ? 0.0f : (m3 * d + b3);
    float f   = (flag != 0) ? 1.0f : 0.0f;

    float v = p * (uTP * f + u_fp * (1.0f - f)) + (1.0f - p) * (uFN * f);
    return in_win ? v : 0.0f;   // all four utilities are 0 outside the window
}

// wave32 shuffle reduction + single LDS exchange among the 8 wave leaders.
// Result valid in thread 0. Fixed order -> deterministic.
__device__ __forceinline__ double block_reduce(double v) {
#pragma unroll
    for (int off = 16; off > 0; off >>= 1)
        v += __shfl_down(v, off, 32);

    __shared__ double sm[WAVES];
    const int lane = threadIdx.x & 31;
    const int wave = threadIdx.x >> 5;
    if (lane == 0) sm[wave] = v;
    __syncthreads();

    if (wave == 0) {
        v = (lane < WAVES) ? sm[lane] : 0.0;
#pragma unroll
        for (int off = WAVES / 2; off > 0; off >>= 1)
            v += __shfl_down(v, off, 32);
    }
    return v;
}

__global__ __launch_bounds__(BLOCK)
void util_loss_partial(const float* __restrict__ pred,
                       const int*   __restrict__ flag,
                       const float* __restrict__ ts,
                       const float* __restrict__ ct,
                       double* __restrict__ partial,
                       int n) {
    const int tid    = blockIdx.x * BLOCK + threadIdx.x;
    const int stride = gridDim.x * BLOCK;
    const int n8     = n >> 3;   // 8 floats (two vec4) per thread per iteration

    const v4f* p4 = (const v4f*)pred;
    const v4i* f4 = (const v4i*)flag;
    const v4f* s4 = (const v4f*)ts;
    const v4f* c4 = (const v4f*)ct;

    double acc = 0.0;

    for (int i = tid; i < n8; i += stride) {
        const int j = i << 1;
        const int jn = (i + stride) << 1;
        // Speculative prefetches (ISA: translation failure silently dropped)
        // -> no bounds guard, no exec-mask churn in the hot loop.
        __builtin_prefetch(p4 + jn, 0, 0);
        __builtin_prefetch(f4 + jn, 0, 0);
        __builtin_prefetch(s4 + jn, 0, 0);
        __builtin_prefetch(c4 + jn, 0, 0);

        // Single-use stream: non-temporal 128-bit loads, 8 in flight.
        v4f p0 = __builtin_nontemporal_load(p4 + j);
        v4f p1 = __builtin_nontemporal_load(p4 + j + 1);
        v4i f0 = __builtin_nontemporal_load(f4 + j);
        v4i f1 = __builtin_nontemporal_load(f4 + j + 1);
        v4f s0 = __builtin_nontemporal_load(s4 + j);
        v4f s1 = __builtin_nontemporal_load(s4 + j + 1);
        v4f c0 = __builtin_nontemporal_load(c4 + j);
        v4f c1 = __builtin_nontemporal_load(c4 + j + 1);

        float sum8 = 0.0f;
#pragma unroll
        for (int k = 0; k < 4; ++k) {
            sum8 += u_elem(p0[k], f0[k], s0[k], c0[k]);
            sum8 += u_elem(p1[k], f1[k], s1[k], c1[k]);
        }
        acc += (double)sum8;   // f64 accumulation across 16.7M O(1) terms
    }

    // Scalar tail (N = 2^24 here, but stay general).
    for (int i = (n8 << 3) + tid; i < n; i += stride)
        acc += (double)u_elem(pred[i], flag[i], ts[i], ct[i]);

    double r = block_reduce(acc);
    if (threadIdx.x == 0) partial[blockIdx.x] = r;
}

__global__ __launch_bounds__(BLOCK)
void util_loss_final(const double* __restrict__ partial,
                     float* __restrict__ out,
                     int npart) {
    double acc = 0.0;
    for (int i = threadIdx.x; i < npart; i += BLOCK)
        acc += partial[i];                      // fixed order -> deterministic
    double r = block_reduce(acc);
    if (threadIdx.x == 0) out[0] = (float)(-r); // loss = sum(-u_func)
}

extern "C" void kernel_launch(void* const* d_in, const int* in_sizes, int n_in,
                              void* d_out, int out_size, void* d_ws, size_t ws_size,
                              hipStream_t stream) {
    // setup_inputs() order:
    //   [0] predictions f32, [1] targets f32 (UNUSED by reference),
    //   [2] sepsis_flags i32, [3] sepsis_times f32, [4] current_times f32
    const float* pred = (const float*)d_in[0];
    const int*   flag = (const int*)d_in[2];
    const float* ts   = (const float*)d_in[3];
    const float* ct   = (const float*)d_in[4];
    float* out = (float*)d_out;
    const int n = in_sizes[0];

    // ws_size is constant across calls -> launch config (and result) is
    // deterministic across replays.
    int grid = GRID_MAX;
    int max_part = (int)(ws_size / sizeof(double));
    if (max_part < 1) max_part = 1;
    if (grid > max_part) grid = max_part;

    double* partial = (double*)d_ws;
    util_loss_partial<<<grid, BLOCK, 0, stream>>>(pred, flag, ts, ct, partial, n);
    util_loss_final<<<1, BLOCK, 0, stream>>>(partial, out, grid);
}